// ODEFunc_25185688224003
// MI455X (gfx1250) — compile-verified
//
#include <hip/hip_runtime.h>
#include <hip/hip_bf16.h>

typedef __attribute__((ext_vector_type(2))) float v2f;
typedef __attribute__((ext_vector_type(8))) float v8f;

#define DIM 128
#define LN_EPS 1e-5f

// ---------------------------------------------------------------------------
// k_init: zero the whole output buffer (dh region doubles as the agg
// accumulator; tail covers the zeros_like outputs) and set deg[i] = 1.0f
// (self-loop contribution to the degree).
// ---------------------------------------------------------------------------
__global__ void k_init(float* __restrict__ out, long long n_out,
                       float* __restrict__ deg, int n_nodes) {
    long long i = (long long)blockIdx.x * blockDim.x + threadIdx.x;
    if (i < n_out) out[i] = 0.0f;
    if (i < n_nodes) deg[i] = 1.0f;
}

// ---------------------------------------------------------------------------
// k_deg: deg[dst[e]] += 1 for every edge (f32 global atomics).
// ---------------------------------------------------------------------------
__global__ void k_deg(const long long* __restrict__ dst,
                      float* __restrict__ deg, long long n_edges) {
    long long e = (long long)blockIdx.x * blockDim.x + threadIdx.x;
    if (e < n_edges) atomicAdd(&deg[dst[e]], 1.0f);
}

// ---------------------------------------------------------------------------
// k_dinv: deg -> rsqrt(deg) in place (deg >= 1 always, self-loops).
// ---------------------------------------------------------------------------
__global__ void k_dinv(float* __restrict__ deg, int n_nodes) {
    int i = blockIdx.x * blockDim.x + threadIdx.x;
    if (i < n_nodes) deg[i] = rsqrtf(deg[i]);
}

// ---------------------------------------------------------------------------
// k_gemm: xw = h @ W via V_WMMA_F32_16X16X4_F32 (fp32 matrix core path).
// One wave computes one 16x16 tile of C. blockDim = 256 (8 waves); wave w
// handles column-tile w (DIM/16 == 8); blockIdx.x is the row-tile.
//
// Fragment layouts (CDNA5 ISA 7.12.2):
//   A (16x4 f32):  lanes 0-15 -> M=lane, VGPR0=K0,VGPR1=K1
//                  lanes 16-31 -> M=lane-16, VGPR0=K2,VGPR1=K3
//   B (4x16 f32):  VGPR v, lanes 0-15 -> row v,   N=lane
//                            lanes 16-31 -> row v+2, N=lane-16
//   C (16x16 f32): VGPR r, lanes 0-15 -> M=r, lanes 16-31 -> M=r+8
// ---------------------------------------------------------------------------
__global__ void k_gemm(const float* __restrict__ h, const float* __restrict__ W,
                       float* __restrict__ xw) {
    const int wave  = threadIdx.x >> 5;      // 0..7 -> column tile
    const int lane  = threadIdx.x & 31;
    const int row0  = blockIdx.x * 16;
    const int col0  = wave * 16;
    const int mrow  = lane & 15;
    const int khalf = (lane >> 4) << 1;      // 0 for lanes 0-15, 2 for 16-31

    const float* hrow = h + (size_t)(row0 + mrow) * DIM;
    const int    col  = col0 + mrow;

    v8f acc = {};
    #pragma unroll
    for (int kk = 0; kk < DIM; kk += 4) {
        v2f a, b;
        a.x = hrow[kk + khalf];
        a.y = hrow[kk + khalf + 1];
        b.x = W[(size_t)(kk + khalf)     * DIM + col];
        b.y = W[(size_t)(kk + khalf + 1) * DIM + col];
        acc = __builtin_amdgcn_wmma_f32_16x16x4_f32(
            /*neg_a=*/false, a, /*neg_b=*/false, b,
            /*c_mod=*/(short)0, acc, /*reuse_a=*/false, /*reuse_b=*/false);
    }

    const int rbase = row0 + ((lane >> 4) << 3);   // +8 rows for high lanes
    #pragma unroll
    for (int r = 0; r < 8; ++r)
        xw[(size_t)(rbase + r) * DIM + col] = acc[r];
}

// Scalar fallback for any rows beyond the last full 16-row tile.
__global__ void k_gemm_rem(const float* __restrict__ h, const float* __restrict__ W,
                           float* __restrict__ xw, int row_start, int n_rows) {
    int t   = blockIdx.x * blockDim.x + threadIdx.x;
    int row = row_start + t / DIM;
    int col = t % DIM;
    if (row >= n_rows) return;
    float s = 0.0f;
    for (int k = 0; k < DIM; ++k)
        s += h[(size_t)row * DIM + k] * W[(size_t)k * DIM + col];
    xw[(size_t)row * DIM + col] = s;
}

// ---------------------------------------------------------------------------
// k_scatter: one wave per edge; each lane handles 4 consecutive dims.
// agg[dst] += xw[src] * dinv[src] * dinv[dst]   (f32 global atomics into the
// dh region of d_out).
// ---------------------------------------------------------------------------
__global__ void k_scatter(const long long* __restrict__ src,
                          const long long* __restrict__ dst,
                          const float* __restrict__ xw,
                          const float* __restrict__ dinv,
                          float* __restrict__ agg, long long n_edges) {
    long long t = (long long)blockIdx.x * blockDim.x + threadIdx.x;
    long long e = t >> 5;
    if (e >= n_edges) return;
    const int lane = (int)(t & 31);

    const long long s = src[e];
    const long long d = dst[e];
    const float norm = dinv[s] * dinv[d];

    const float4 m = *((const float4*)(xw + s * DIM) + lane);
    float* ad = agg + d * DIM + lane * 4;
    atomicAdd(ad + 0, m.x * norm);
    atomicAdd(ad + 1, m.y * norm);
    atomicAdd(ad + 2, m.z * norm);
    atomicAdd(ad + 3, m.w * norm);
}

// ---------------------------------------------------------------------------
// k_final: one wave per node (8 nodes per 256-thread block). Adds the
// self-loop term and bias, does LayerNorm over 128 dims via wave32
// shuffle-reduction, applies gamma/beta and tanh, writes dh in place.
// ---------------------------------------------------------------------------
__global__ void k_final(float* __restrict__ out, const float* __restrict__ xw,
                        const float* __restrict__ dinv,
                        const float* __restrict__ bias,
                        const float* __restrict__ gamma,
                        const float* __restrict__ beta, int n_nodes) {
    const int wave = threadIdx.x >> 5;
    const int lane = threadIdx.x & 31;
    const int node = blockIdx.x * 8 + wave;
    if (node >= n_nodes) return;

    const float di   = dinv[node];
    const float self = di * di;
    const size_t base = (size_t)node * DIM + lane * 4;

    float v[4];
    float s = 0.0f, s2 = 0.0f;
    #pragma unroll
    for (int j = 0; j < 4; ++j) {
        const int d = lane * 4 + j;
        float val = out[base + j] + xw[base + j] * self + bias[d];
        v[j] = val;
        s  += val;
        s2 += val * val;
    }
    // wave32 butterfly reduction for sum and sum-of-squares
    #pragma unroll
    for (int off = 16; off > 0; off >>= 1) {
        s  += __shfl_xor(s,  off, 32);
        s2 += __shfl_xor(s2, off, 32);
    }
    const float mean = s * (1.0f / DIM);
    const float var  = s2 * (1.0f / DIM) - mean * mean;
    const float rstd = rsqrtf(var + LN_EPS);

    #pragma unroll
    for (int j = 0; j < 4; ++j) {
        const int d = lane * 4 + j;
        float o = (v[j] - mean) * rstd * gamma[d] + beta[d];
        out[base + j] = tanhf(o);
    }
}

// ---------------------------------------------------------------------------
// kernel_launch
// inputs (setup_inputs order): t, h, edge_index(int64 2xE), batch_size,
//                              W, b, gamma, beta
// ---------------------------------------------------------------------------
extern "C" void kernel_launch(void* const* d_in, const int* in_sizes, int n_in,
                              void* d_out, int out_size, void* d_ws, size_t ws_size,
                              hipStream_t stream) {
    const float*     h    = (const float*)d_in[1];
    const long long* ei   = (const long long*)d_in[2];
    const float*     W    = (const float*)d_in[4];
    const float*     bias = (const float*)d_in[5];
    const float*     gam  = (const float*)d_in[6];
    const float*     bet  = (const float*)d_in[7];

    const int       n_nodes = in_sizes[1] / DIM;        // 50000
    const long long n_edges = (long long)in_sizes[2] / 2; // 800000
    const long long* src = ei;
    const long long* dst = ei + n_edges;

    // workspace layout: [deg/dinv: n_nodes floats][xw: n_nodes*DIM floats]
    float* deg = (float*)d_ws;
    size_t xw_off = (((size_t)n_nodes * sizeof(float)) + 255) & ~(size_t)255;
    float* xw = (float*)((char*)d_ws + xw_off);

    float* out = (float*)d_out;

    const int BLK = 256;

    // 1) zero output (agg region + zero-outputs tail), deg = 1 (self-loop)
    {
        long long total = out_size > n_nodes ? (long long)out_size : (long long)n_nodes;
        unsigned int blocks = (unsigned int)((total + BLK - 1) / BLK);
        k_init<<<blocks, BLK, 0, stream>>>(out, (long long)out_size, deg, n_nodes);
    }
    // 2) degree accumulation over edges
    {
        unsigned int blocks = (unsigned int)((n_edges + BLK - 1) / BLK);
        k_deg<<<blocks, BLK, 0, stream>>>(dst, deg, n_edges);
    }
    // 3) dinv = rsqrt(deg)
    {
        unsigned int blocks = (unsigned int)((n_nodes + BLK - 1) / BLK);
        k_dinv<<<blocks, BLK, 0, stream>>>(deg, n_nodes);
    }
    // 4) xw = h @ W  (fp32 WMMA)
    {
        int m_tiles = n_nodes / 16;
        if (m_tiles > 0)
            k_gemm<<<m_tiles, BLK, 0, stream>>>(h, W, xw);
        int rem_start = m_tiles * 16;
        int rem_rows  = n_nodes - rem_start;
        if (rem_rows > 0) {
            unsigned int blocks = (unsigned int)(((long long)rem_rows * DIM + BLK - 1) / BLK);
            k_gemm_rem<<<blocks, BLK, 0, stream>>>(h, W, xw, rem_start, n_nodes);
        }
    }
    // 5) edge scatter-add into out (agg)
    {
        long long threads = n_edges * 32;
        unsigned int blocks = (unsigned int)((threads + BLK - 1) / BLK);
        k_scatter<<<blocks, BLK, 0, stream>>>(src, dst, xw, deg, out, n_edges);
    }
    // 6) self-loop + bias + LayerNorm + tanh, in place
    {
        unsigned int blocks = (unsigned int)((n_nodes + 7) / 8);
        k_final<<<blocks, BLK, 0, stream>>>(out, xw, deg, bias, gam, bet, n_nodes);
    }
}